// FilterBank_38938173505900
// MI455X (gfx1250) — compile-verified
//
#include <hip/hip_runtime.h>

// CDNA5 / gfx1250 FIR bandpass as banded-Toeplitz WMMA (fp32 matrix pipe).
// y[s,t] = sum_k h[k] * xe(s, t+206-k), 413 taps, reflect-limited edges.
//
// Each wave: 2 M-tiles (time) x 2 N-tiles (signals) = 4 independent
// accumulator chains; A (taps) shared across N-tiles, B (signals) shared
// across M-tiles via the zero-padded reversed tap array (union-band trick).

typedef __attribute__((ext_vector_type(2))) float v2f;
typedef __attribute__((ext_vector_type(8))) float v8f;

namespace {
constexpr int kT       = 4000;               // samples per signal
constexpr int kSig     = 64 * 64;            // independent signals
constexpr int kTaps    = 413;
constexpr int kHalf    = (kTaps - 1) / 2;    // 206
constexpr int kBlkS    = 64;                 // signals per workgroup
constexpr int kBlkT    = 128;                // outputs per workgroup
constexpr int kWin     = kBlkT + 2 * kHalf;  // 540 staged inputs per signal
constexpr int kStride  = 546;                // LDS row stride (34 mod 64: conflict-free)
constexpr int kKSteps  = 111;                // union band of 2 M-tiles: (413+15+16)/4
constexpr int kHpLen   = 480;                // reversed taps + zero pads both sides
constexpr int kHpOff   = 32;                 // left zero pad (covers the -16 shifted tile)
}

__global__ __launch_bounds__(256)
void fir_bandpass_wmma2x2(const float* __restrict__ x,
                          const float* __restrict__ h,
                          float* __restrict__ out)
{
    // 64*546*4 = 139,776 B window + 2*480*4 = 3,840 B taps = 143.6 KB
    // (CDNA5 allows up to 320 KB LDS per workgroup).
    __shared__ float xs[kBlkS * kStride];
    __shared__ float hpA[kHpLen];   // copy for lanes 0-15
    __shared__ float hpB[kHpLen];   // +32-bank shifted copy for lanes 16-31

    const int tid = threadIdx.x;
    const int s0b = blockIdx.x * kBlkS;   // first signal of this block
    const int t0b = blockIdx.y * kBlkT;   // first output time of this block

    // ---- stage reversed + zero-padded taps (two bank-shifted copies) ----
    for (int i = tid; i < kHpLen; i += 256) {
        int j = i - kHpOff;
        float v = (j >= 0 && j < kTaps) ? h[kTaps - 1 - j] : 0.0f;
        hpA[i] = v;
        hpB[i] = v;
    }

    // ---- stage input window with reflect-limited padding ----
    for (int s = (tid >> 4); s < kBlkS; s += 16) {
        const float* row = x + (size_t)(s0b + s) * kT;
        const float xf = row[0];
        const float xl = row[kT - 1];
        for (int jj = (tid & 15); jj < kWin; jj += 16) {
            int j = t0b - kHalf + jj;                 // global input time
            float v;
            if (j < 0)        v = 2.0f * xf - row[-j];
            else if (j >= kT) v = 2.0f * xl - row[2 * kT - 2 - j];
            else              v = row[j];
            xs[s * kStride + jj] = v;
        }
    }
    __syncthreads();

    const int w    = tid >> 5;            // wave 0..7
    const int lane = tid & 31;
    const int mg   = w & 3;               // M-group: times t0b + 32*mg + {0..31}
    const int ng   = w >> 2;              // N-group: signals s0b + 32*ng + {0..31}
    const int t00  = t0b + (mg << 5);     // first M-tile base; second is t00+16
    if (t00 >= kT) return;                // wave-uniform (kT % 32 == 0)

    const int m  = lane & 15;             // A row (time) == B col (signal)
    const int hi = lane >> 4;             // half-wave
    const int kk = hi << 1;               // K slot held by this lane (0 or 2)

    // A0[m][kk+c] = h[412 - 4k + m - kk - c] = hp[kHpOff + 4k - m + kk + c]
    // A1 (tile t00+16) is the same stream shifted by -16 taps.
    const float* hb  = (hi ? hpB : hpA) + (kHpOff + kk - m);
    // B rows: N0 -> xs[32*ng + m], N1 -> +16 rows; col = 32*mg + 4k + kk.
    const float* xb0 = xs + ((ng << 5) + m) * kStride + (mg << 5) + kk;
    const float* xb1 = xb0 + 16 * kStride;

    v8f acc00 = {0.f,0.f,0.f,0.f,0.f,0.f,0.f,0.f};   // (t00,    N0)
    v8f acc01 = acc00;                               // (t00,    N1)
    v8f acc10 = acc00;                               // (t00+16, N0)
    v8f acc11 = acc00;                               // (t00+16, N1)

    #pragma unroll 4
    for (int k = 0; k < kKSteps; ++k) {
        const int c = 4 * k;
        v2f a0, a1, b0, b1;
        a0.x = hb[c];          a0.y = hb[c + 1];        // taps, tile t00
        a1.x = hb[c - 16];     a1.y = hb[c - 15];       // taps, tile t00+16
        b0.x = xb0[c];         b0.y = xb0[c + 1];       // signals, N0
        b1.x = xb1[c];         b1.y = xb1[c + 1];       // signals, N1
        acc00 = __builtin_amdgcn_wmma_f32_16x16x4_f32(
            false, a0, false, b0, (short)0, acc00, false, false);
        acc01 = __builtin_amdgcn_wmma_f32_16x16x4_f32(
            false, a0, false, b1, (short)0, acc01, false, false);
        acc10 = __builtin_amdgcn_wmma_f32_16x16x4_f32(
            false, a1, false, b0, (short)0, acc10, false, false);
        acc11 = __builtin_amdgcn_wmma_f32_16x16x4_f32(
            false, a1, false, b1, (short)0, acc11, false, false);
    }

    // D layout: VGPR r -> time t0 + r + 8*hi, lane -> signal s0 + m.
    // Each lane owns 8 consecutive samples per tile -> aligned float4 stores.
    float* o00 = out + (size_t)(s0b + (ng << 5) + m) * kT + t00 + (hi << 3);
    float* o01 = o00 + (size_t)16 * kT;     // N1: +16 signals
    float* o10 = o00 + 16;                  // M1: +16 samples
    float* o11 = o01 + 16;

    float4 v;
    v.x = acc00[0]; v.y = acc00[1]; v.z = acc00[2]; v.w = acc00[3];
    *(float4*)(o00) = v;
    v.x = acc00[4]; v.y = acc00[5]; v.z = acc00[6]; v.w = acc00[7];
    *(float4*)(o00 + 4) = v;
    v.x = acc01[0]; v.y = acc01[1]; v.z = acc01[2]; v.w = acc01[3];
    *(float4*)(o01) = v;
    v.x = acc01[4]; v.y = acc01[5]; v.z = acc01[6]; v.w = acc01[7];
    *(float4*)(o01 + 4) = v;
    v.x = acc10[0]; v.y = acc10[1]; v.z = acc10[2]; v.w = acc10[3];
    *(float4*)(o10) = v;
    v.x = acc10[4]; v.y = acc10[5]; v.z = acc10[6]; v.w = acc10[7];
    *(float4*)(o10 + 4) = v;
    v.x = acc11[0]; v.y = acc11[1]; v.z = acc11[2]; v.w = acc11[3];
    *(float4*)(o11) = v;
    v.x = acc11[4]; v.y = acc11[5]; v.z = acc11[6]; v.w = acc11[7];
    *(float4*)(o11 + 4) = v;
}

extern "C" void kernel_launch(void* const* d_in, const int* in_sizes, int n_in,
                              void* d_out, int out_size, void* d_ws, size_t ws_size,
                              hipStream_t stream) {
    const float* x = (const float*)d_in[0];   // [64, 64, 4000] f32
    const float* h = (const float*)d_in[1];   // [413] f32
    float* out = (float*)d_out;               // [64, 64, 4000] f32

    dim3 grid(kSig / kBlkS, (kT + kBlkT - 1) / kBlkT);  // 64 x 32 workgroups
    fir_bandpass_wmma2x2<<<grid, 256, 0, stream>>>(x, h, out);
}